// CustomLayer_39625368273011
// MI455X (gfx1250) — compile-verified
//
#include <hip/hip_runtime.h>
#include <stdint.h>

#define AS1 __attribute__((address_space(1)))
#define AS3 __attribute__((address_space(3)))

#ifndef __has_builtin
#define __has_builtin(x) 0
#endif

#if __has_builtin(__builtin_amdgcn_global_load_async_to_lds_b32) && \
    __has_builtin(__builtin_amdgcn_s_wait_asynccnt)
#define USE_ASYNC_LDS 1
#else
#define USE_ASYNC_LDS 0
#endif

#if USE_ASYNC_LDS
#define ASYNC_WAIT(n)                         \
  do {                                        \
    __builtin_amdgcn_s_wait_asynccnt(n);      \
    __asm__ __volatile__("" ::: "memory");    \
  } while (0)
#else
#define ASYNC_WAIT(n) __asm__ __volatile__("" ::: "memory")
#endif

static constexpr int MROWS = 4096;
static constexpr int NCOLS = 4096;
static constexpr int BATCH = 64;          // dense batch of vectors
static constexpr int ROWS_PER_BLOCK = 8;  // 8 wave32 per workgroup, one row each
static constexpr int CHUNK = 32;          // nonzeros staged per wave per step

struct Pair {
  float v;  // nonzero value
  int c;    // column index
};

// ---------------------------------------------------------------------------
// Main CSR SpMM: Y[64x4096] = A_csr * X. One wave32 per row. Each lane owns
// batch columns (2*lane, 2*lane+1) as a float2 accumulator.
//
// The (val,col) streams are staged into LDS in chunks of 32 nonzeros with
// CDNA5 async global->LDS copies (ASYNCcnt-tracked, double buffered,
// s_wait_asynccnt-synchronized). The two per-lane b32 copies land interleaved
// so each nonzero is consumed with a single ds_load_b64 broadcast. The x-row
// gather is one fully coalesced 256B float2 load per nonzero per wave and
// stays L2-resident (X fits in 1MB of the 192MB L2).
// ---------------------------------------------------------------------------
__global__ __launch_bounds__(256) void spmm_csr_kernel(
    const float* __restrict__ vals, const int* __restrict__ cols,
    const int* __restrict__ rowptr, const float* __restrict__ xT,
    float* __restrict__ y) {
  __shared__ Pair s_pair[ROWS_PER_BLOCK][2][CHUNK];

  const int wave = threadIdx.x >> 5;
  const int lane = threadIdx.x & 31;
  const int row = blockIdx.x * ROWS_PER_BLOCK + wave;

  const int start = rowptr[row];
  const int end = rowptr[row + 1];
  const int len = end - start;
  const int nfull = len >> 5;  // full chunks of 32 nonzeros

  const float2* __restrict__ xT2 = (const float2*)xT;
  float2 acc = make_float2(0.0f, 0.0f);

  auto issue = [&](int buf, int base) {
#if USE_ASYNC_LDS
    // Per-lane 4B async copies; 32 lanes cover 128B of each stream per
    // instruction. LDS destinations are interleaved (stride 8) so the
    // consumer reads one b64 per nonzero.
    __builtin_amdgcn_global_load_async_to_lds_b32(
        (AS1 int*)(unsigned long long)(vals + base + lane),
        (AS3 int*)(unsigned)(unsigned long long)(&s_pair[wave][buf][lane].v),
        0, 0);
    __builtin_amdgcn_global_load_async_to_lds_b32(
        (AS1 int*)(unsigned long long)(cols + base + lane),
        (AS3 int*)(unsigned)(unsigned long long)(&s_pair[wave][buf][lane].c),
        0, 0);
#else
    s_pair[wave][buf][lane].v = vals[base + lane];
    s_pair[wave][buf][lane].c = cols[base + lane];
#endif
  };

  if (nfull > 0) {
    issue(0, start);  // prologue: chunk 0 into buffer 0
    for (int ci = 0; ci < nfull; ++ci) {
      const int buf = ci & 1;
      if (ci + 1 < nfull) {
        issue(buf ^ 1, start + (ci + 1) * CHUNK);  // prefetch next chunk
        ASYNC_WAIT(2);  // 4 outstanding; <=2 left => current chunk landed
      } else {
        ASYNC_WAIT(0);
      }
#pragma unroll 4
      for (int j = 0; j < CHUNK; ++j) {
        const Pair p = s_pair[wave][buf][j];     // one ds_load_b64 broadcast
        const float2 xv = xT2[(p.c << 5) + lane];  // coalesced 256B/wave
        acc.x = fmaf(p.v, xv.x, acc.x);
        acc.y = fmaf(p.v, xv.y, acc.y);
      }
    }
  }

  // Remainder nonzeros (< 32): plain broadcast loads.
  for (int k = start + nfull * CHUNK; k < end; ++k) {
    const float v = vals[k];
    const int c = cols[k];
    const float2 xv = xT2[(c << 5) + lane];
    acc.x = fmaf(v, xv.x, acc.x);
    acc.y = fmaf(v, xv.y, acc.y);
  }

  // Output layout [B, M]: lane holds batches 2*lane and 2*lane+1.
  y[(2 * lane) * MROWS + row] = acc.x;
  y[(2 * lane + 1) * MROWS + row] = acc.y;
}

// ---------------------------------------------------------------------------
// Transpose x[64][4096] (row-major) -> xT[4096][64] so that the per-nonzero
// gather of x[:, col] becomes one contiguous 256B line. LDS-tiled, 33-stride
// padding keeps both LDS phases bank-conflict free on 64 banks.
// ---------------------------------------------------------------------------
__global__ __launch_bounds__(256) void spmm_xpose_kernel(
    const float* __restrict__ x, float* __restrict__ xT) {
  __shared__ float tile[32][33];
  const int n0 = blockIdx.x * 32;
  const int b0 = blockIdx.y * 32;
  const int tx = threadIdx.x;  // 0..31
  const int ty = threadIdx.y;  // 0..7
#pragma unroll
  for (int i = ty; i < 32; i += 8)
    tile[i][tx] = x[(b0 + i) * NCOLS + n0 + tx];
  __syncthreads();
#pragma unroll
  for (int i = ty; i < 32; i += 8)
    xT[(n0 + i) * BATCH + b0 + tx] = tile[tx][i];
}

extern "C" void kernel_launch(void* const* d_in, const int* in_sizes, int n_in,
                              void* d_out, int out_size, void* d_ws,
                              size_t ws_size, hipStream_t stream) {
  (void)in_sizes;
  (void)n_in;
  (void)out_size;
  (void)ws_size;
  const float* x = (const float*)d_in[0];     // [64, 4096]
  const float* vals = (const float*)d_in[1];  // [nnz]
  const int* cols = (const int*)d_in[2];      // [nnz]
  const int* rowptr = (const int*)d_in[3];    // [4097]
  float* xT = (float*)d_ws;                   // [4096, 64] = 1 MB scratch
  float* y = (float*)d_out;                   // [64, 4096]

  dim3 tb(32, 8);
  dim3 tg(NCOLS / 32, BATCH / 32);
  spmm_xpose_kernel<<<tg, tb, 0, stream>>>(x, xT);
  spmm_csr_kernel<<<MROWS / ROWS_PER_BLOCK, ROWS_PER_BLOCK * 32, 0, stream>>>(
      vals, cols, rowptr, xT, y);
}